// AUC_like_loss_43843026157975
// MI455X (gfx1250) — compile-verified
//
#include <hip/hip_runtime.h>

// Problem constants (match reference: N = 16384, f32 in/out).
#define PN 16384
#define ROWS_PER_BLOCK 8
#define THREADS 256
#define NUM_BLOCKS (PN / ROWS_PER_BLOCK)   // 2048 blocks, one f32 partial each
#define JITERS (PN / (THREADS * 2))        // 32 iterations, two j per thread per iter

typedef float v2f __attribute__((ext_vector_type(2)));
typedef float v8f __attribute__((ext_vector_type(8)));

// CDNA5 has V_TANH_F32 in its transcendental set: sigmoid(x) = 0.5 + 0.5*tanh(x/2)
// costs ONE trans op instead of exp+rcp (two). Guarded so the kernel always compiles.
#if __has_builtin(__builtin_amdgcn_tanhf)
  #define HAVE_TANH 1
  #define FAST_TANHF __builtin_amdgcn_tanhf
#elif __has_builtin(__builtin_amdgcn_tanh_f32)
  #define HAVE_TANH 1
  #define FAST_TANHF __builtin_amdgcn_tanh_f32
#else
  #define HAVE_TANH 0
#endif

__device__ __forceinline__ float sigmoid_from_scaled(float xs) {
#if HAVE_TANH
    // xs = 0.5 * p_i * p_j
    return fmaf(0.5f, FAST_TANHF(xs), 0.5f);
#else
    // xs = -log2(e) * p_i * p_j ;  sigma = 1/(1+exp2(xs))
    return __builtin_amdgcn_rcpf(1.0f + __builtin_amdgcn_exp2f(xs));
#endif
}

// Kernel 1: each block owns 8 rows i (block-uniform -> SGPR-resident constants),
// threads cover j with uniform trip count and float2 (b64) coalesced loads.
// Four f32 accumulators feed a V_WMMA_F32_16X16X4_F32 all-ones reduction.
__global__ __launch_bounds__(THREADS)
void auc_pair_partial(const float* __restrict__ y_true,
                      const float* __restrict__ y_pred,
                      float* __restrict__ partials) {
    const int tid  = threadIdx.x;
    const int row0 = blockIdx.x * ROWS_PER_BLOCK;

    // Block-uniform row constants (scalar loads).
    float q[ROWS_PER_BLOCK];   // pre-scaled p_i for the sigmoid path in use
    float ti[ROWS_PER_BLOCK];
#pragma unroll
    for (int r = 0; r < ROWS_PER_BLOCK; ++r) {
#if HAVE_TANH
        q[r] = 0.5f * y_pred[row0 + r];
#else
        q[r] = -1.4426950408889634f * y_pred[row0 + r];
#endif
        ti[r] = y_true[row0 + r];
    }

    float acc0 = 0.0f, acc1 = 0.0f, acc2 = 0.0f, acc3 = 0.0f;
    for (int it = 0; it < JITERS; ++it) {
        const int j0 = (it * THREADS + tid) * 2;   // uniform trip count, scalar loop ctrl
        const float2 pj = *(const float2*)(y_pred + j0);
        const float2 tj = *(const float2*)(y_true + j0);
#pragma unroll
        for (int r = 0; r < ROWS_PER_BLOCK; ++r) {
            const float sg0 = sigmoid_from_scaled(q[r] * pj.x);
            const float sg1 = sigmoid_from_scaled(q[r] * pj.y);
            const float m0  = fmaxf(ti[r] - tj.x, 0.0f);
            const float m1  = fmaxf(ti[r] - tj.y, 0.0f);
            // Spread FMAs over 4 accumulators for ILP (4 per acc per iteration).
            if (r & 1) { acc2 = fmaf(sg0, m0, acc2); acc3 = fmaf(sg1, m1, acc3); }
            else       { acc0 = fmaf(sg0, m0, acc0); acc1 = fmaf(sg1, m1, acc1); }
        }
    }

    // --- Wave-level reduction via WMMA (f32, exact adds) ---
    // D = ones(16x4) x B(4x16): every row of D is the vector of B's column sums;
    // summing row 0 across lanes 0..15 totals all 64 per-wave partials.
    // Layout-proof: with A == all-ones, each lane's two B values enter the total once.
    v2f A; A.x = 1.0f;        A.y = 1.0f;
    v2f B; B.x = acc0 + acc1; B.y = acc2 + acc3;
    v8f C = {};
    C = __builtin_amdgcn_wmma_f32_16x16x4_f32(
        /*neg_a=*/false, A, /*neg_b=*/false, B,
        /*c_mod=*/(short)0, C, /*reuse_a=*/false, /*reuse_b=*/false);

    __shared__ float red[(THREADS / 32) * 16];  // 8 waves x 16 column sums
    const int lane = tid & 31;
    const int wave = tid >> 5;
    if (lane < 16) red[wave * 16 + lane] = C[0];  // D[M=0][N=lane]
    __syncthreads();

    if (tid == 0) {
        float s = 0.0f;
#pragma unroll
        for (int k = 0; k < (THREADS / 32) * 16; ++k) s += red[k];
        partials[blockIdx.x] = s;
    }
}

// Kernel 2: deterministic final reduce of 2048 block partials, scale by -1/N^2.
__global__ __launch_bounds__(256)
void auc_final_reduce(const float* __restrict__ partials, float* __restrict__ out) {
    __shared__ float sm[256];
    float s = 0.0f;
    for (int k = threadIdx.x; k < NUM_BLOCKS; k += 256) s += partials[k];
    sm[threadIdx.x] = s;
    __syncthreads();
    for (int off = 128; off > 0; off >>= 1) {
        if ((int)threadIdx.x < off) sm[threadIdx.x] += sm[threadIdx.x + off];
        __syncthreads();
    }
    if (threadIdx.x == 0) {
        const float inv = 1.0f / ((float)PN * (float)PN);
        out[0] = -sm[0] * inv;
    }
}

extern "C" void kernel_launch(void* const* d_in, const int* in_sizes, int n_in,
                              void* d_out, int out_size, void* d_ws, size_t ws_size,
                              hipStream_t stream) {
    const float* y_true = (const float*)d_in[0];
    const float* y_pred = (const float*)d_in[1];
    float* partials = (float*)d_ws;   // needs NUM_BLOCKS * 4 = 8 KB of scratch

    auc_pair_partial<<<NUM_BLOCKS, THREADS, 0, stream>>>(y_true, y_pred, partials);
    auc_final_reduce<<<1, 256, 0, stream>>>(partials, (float*)d_out);
}